// DotProductPredictor_47699906789906
// MI455X (gfx1250) — compile-verified
//
#include <hip/hip_runtime.h>
#include <hip/hip_bf16.h>

// Per-edge dot product via V_WMMA_F32_16X16X4_F32.
// Tile = 16 edges per wave32. A = 16 src rows (16xK), B = 16 dst rows (Kx16),
// accumulate K=64 in 16 WMMA steps of K=4; the diagonal of the 16x16 f32
// accumulator is the 16 edge scores.

typedef __attribute__((ext_vector_type(2))) float v2f;
typedef __attribute__((ext_vector_type(8))) float v8f;

#ifndef D_FEAT
#define D_FEAT 64
#endif

__global__ __launch_bounds__(256) void edge_dot_wmma_kernel(
    const float* __restrict__ h,
    const int* __restrict__ src,
    const int* __restrict__ dst,
    float* __restrict__ out,
    int n_edges)
{
    const int lane        = threadIdx.x & 31;
    const int waveInBlock = threadIdx.x >> 5;

    const long long tile     = (long long)blockIdx.x * 8 + waveInBlock;
    const long long tileBase = tile * 16;
    if (tileBase >= n_edges) return;          // wave-uniform exit (EXEC stays all-1s)

    const int m  = lane & 15;                 // edge within tile == output column N
    const int hi = lane >> 4;                 // half-wave selects K pair {2hi, 2hi+1}

    long long e = tileBase + m;
    if (e >= n_edges) e = n_edges - 1;        // clamp so tail lanes load safely

    const int sidx = src[e];
    const int didx = dst[e];

    // A-matrix lane source: row h[sidx], K = k + 2*hi + {0,1}  (8B aligned)
    // B-matrix lane source: row h[didx], same K split (mirrored layout)
    const float* rs = h + (size_t)sidx * D_FEAT + 2 * hi;
    const float* rd = h + (size_t)didx * D_FEAT + 2 * hi;

    v8f acc = {0.f, 0.f, 0.f, 0.f, 0.f, 0.f, 0.f, 0.f};
#pragma unroll
    for (int k = 0; k < D_FEAT; k += 4) {
        v2f a = *(const v2f*)(rs + k);
        v2f b = *(const v2f*)(rd + k);
        // 8 args: (neg_a, A, neg_b, B, c_mod, C, reuse_a, reuse_b)
        acc = __builtin_amdgcn_wmma_f32_16x16x4_f32(
            false, a, false, b, (short)0, acc, false, false);
    }

    // Diagonal extraction: (N,N) lives at lane N (VGPR N) for N<8 and at
    // lane N+16 (VGPR N-8) for N>=8, i.e. active lanes 0-7 and 24-31 with
    // VGPR index r = lane & 7.
    const int r = lane & 7;
    float dval = acc[0];
#pragma unroll
    for (int i = 1; i < 8; ++i)
        if (r == i) dval = acc[i];

    const bool active = (lane < 8) || (lane >= 24);
    const long long eo = tileBase + m;
    if (active && eo < n_edges)
        out[eo] = dval;
}

extern "C" void kernel_launch(void* const* d_in, const int* in_sizes, int n_in,
                              void* d_out, int out_size, void* d_ws, size_t ws_size,
                              hipStream_t stream) {
    const float* h   = (const float*)d_in[0];
    const int*   src = (const int*)d_in[1];
    const int*   dst = (const int*)d_in[2];
    float*       out = (float*)d_out;
    const int E = in_sizes[1];                 // number of edges

    const long long tiles  = ((long long)E + 15) / 16;   // 16 edges per wave
    const int wavesPerBlk  = 8;                          // 256 threads
    const int blocks       = (int)((tiles + wavesPerBlk - 1) / wavesPerBlk);

    edge_dot_wmma_kernel<<<blocks, 256, 0, stream>>>(h, src, dst, out, E);
}